// EdgeNet__13108240188001
// MI455X (gfx1250) — compile-verified
//
#include <hip/hip_runtime.h>

#define NN 2048
#define KK 17
#define PENALTY 10.0f
#define WAVES 4

typedef float v4f __attribute__((ext_vector_type(4)));

__device__ __forceinline__ float wave_sum(float v) {
#pragma unroll
    for (int o = 16; o > 0; o >>= 1) v += __shfl_xor(v, o, 32);
    return v;
}

__global__ __launch_bounds__(WAVES * 32) void edgenet_topk_kernel(
    const float* __restrict__ theta,
    const float* __restrict__ dist,
    const float* __restrict__ ins,   // [2, B*N]
    const float* __restrict__ lW,    // [2,64] flat
    const float* __restrict__ lb,    // [64]
    const float* __restrict__ gW,    // [68]
    const float* __restrict__ gb,    // [1]
    float* __restrict__ out,
    int BN)
{
    __shared__ float shRow[WAVES][NN];                      // staged dist row per wave
    __shared__ unsigned long long shCand[WAVES][32 * KK];   // per-lane sorted top-K lists

    const int lane = threadIdx.x & 31;
    const int wv   = threadIdx.x >> 5;
    const int row  = blockIdx.x * WAVES + wv;
    if (row >= BN) return;

    const float* drow = dist  + (size_t)row * NN;
    const float* trow = theta + (size_t)row * NN;
    float*       orow = out   + (size_t)row * NN;

    // ---- Phase 1: async-stage the dist row into LDS (gfx1250 ASYNC path) ----
    {
        unsigned lds_base = (unsigned)(size_t)(&shRow[wv][0]);
        for (int i = 0; i < 16; ++i) {
            int c4 = i * 32 + lane;                     // 16B chunk index, coalesced
            unsigned lds_addr = lds_base + (unsigned)(c4 * 16);
            unsigned long long gaddr =
                (unsigned long long)(size_t)drow + (unsigned long long)(c4 * 16);
            asm volatile("global_load_async_to_lds_b128 %0, %1, off"
                         :: "v"(lds_addr), "v"(gaddr) : "memory");
        }
        asm volatile("s_wait_asynccnt 0" ::: "memory");
    }

    // ---- Phase 2: per-lane sorted top-17 over 64 elements (key = distbits<<32 | col) ----
    unsigned long long* my = &shCand[wv][lane * KK];
#pragma unroll
    for (int j = 0; j < KK; ++j) my[j] = ~0ull;

    const v4f* srow4 = (const v4f*)(&shRow[wv][0]);
    for (int i = 0; i < 16; ++i) {
        int c4 = i * 32 + lane;
        v4f v = srow4[c4];
        int cbase = c4 * 4;
#pragma unroll
        for (int e = 0; e < 4; ++e) {
            unsigned long long key =
                ((unsigned long long)__float_as_uint(v[e]) << 32) | (unsigned)(cbase + e);
            if (key < my[KK - 1]) {
                int j = KK - 1;
                while (j > 0 && my[j - 1] > key) { my[j] = my[j - 1]; --j; }
                my[j] = key;
            }
        }
    }

    // ---- Phase 3: 17-round 32-way merge via wave-min on u64 keys ----
    int p = 0;
    unsigned long long kept = 0;  // lane t keeps the t-th smallest key
    for (int t = 0; t < KK; ++t) {
        unsigned long long k = (p < KK) ? my[p] : ~0ull;
        unsigned long long m = k;
#pragma unroll
        for (int o = 16; o > 0; o >>= 1) {
            unsigned long long other = __shfl_xor(m, o, 32);
            m = (other < m) ? other : m;
        }
        unsigned long long bal = __ballot(k == m);
        int winner = __ffsll((unsigned long long)bal) - 1;
        if (lane == winner) ++p;
        if (lane == t) kept = m;
    }

    // ---- Phase 4: means + folded-weight scalar math ----
    int   col  = (int)(kept & 0xffffffffu);
    float dval = __uint_as_float((unsigned)(kept >> 32));
    float tval = 0.f, dvs = 0.f;
    if (lane < KK) { tval = trow[col]; dvs = dval; }
    float mean_t = wave_sum(tval) * (1.0f / KK);
    float mean_d = wave_sum(dvs)  * (1.0f / KK);

    // folded weights: w2x = sum_e lW[x][e]*gW[2+e], pb = sum_e lb[e]*gW[2+e]
    float g0 = gW[2 + lane], g1 = gW[2 + lane + 32];
    float w20 = wave_sum(lW[lane]      * g0 + lW[lane + 32]      * g1);
    float w21 = wave_sum(lW[64 + lane] * g0 + lW[64 + lane + 32] * g1);
    float pb  = wave_sum(lb[lane]      * g0 + lb[lane + 32]      * g1);

    float ins0 = ins[row];
    float ins1 = ins[BN + row];
    float A  = gW[0];
    float B1 = gW[1] - 1.0f;
    float C  = mean_t * w20 + mean_d * w21 + pb
             + ins0 * gW[66] + ins1 * gW[67] + gb[0];

    // ---- Phase 5: broadcast 17 (col,val) pairs through LDS, then stream the row ----
    unsigned long long* sel = &shCand[wv][0];  // merge done; reuse region
    if (lane < KK) {
        float val = A * tval + B1 * dval + C;
        sel[lane] = ((unsigned long long)__float_as_uint(val) << 32) | (unsigned)col;
    }

    unsigned scol[KK];
    float    sval[KK];
#pragma unroll
    for (int k = 0; k < KK; ++k) {
        unsigned long long s = sel[k];
        scol[k] = (unsigned)(s & 0xffffffffu);
        sval[k] = __uint_as_float((unsigned)(s >> 32));
    }

    v4f* orow4 = (v4f*)orow;
    for (int i = 0; i < 16; ++i) {
        int c4 = i * 32 + lane;
        v4f v = { PENALTY, PENALTY, PENALTY, PENALTY };
#pragma unroll
        for (int k = 0; k < KK; ++k) {
            if ((scol[k] >> 2) == (unsigned)c4) {   // rare hit: 17/512 chunks
                unsigned d = scol[k] & 3u;
                v[0] = (d == 0u) ? sval[k] : v[0];
                v[1] = (d == 1u) ? sval[k] : v[1];
                v[2] = (d == 2u) ? sval[k] : v[2];
                v[3] = (d == 3u) ? sval[k] : v[3];
            }
        }
        __builtin_nontemporal_store(v, orow4 + c4);  // streaming 128MB output
    }
}

extern "C" void kernel_launch(void* const* d_in, const int* in_sizes, int n_in,
                              void* d_out, int out_size, void* d_ws, size_t ws_size,
                              hipStream_t stream) {
    const float* theta = (const float*)d_in[0];
    const float* dist  = (const float*)d_in[1];
    const float* ins   = (const float*)d_in[2];
    const float* lW    = (const float*)d_in[3];
    const float* lb    = (const float*)d_in[4];
    const float* gW    = (const float*)d_in[5];
    const float* gb    = (const float*)d_in[6];

    const int BN = 8 * 2048;                    // B*N rows
    const int blocks = (BN + WAVES - 1) / WAVES;
    edgenet_topk_kernel<<<blocks, WAVES * 32, 0, stream>>>(
        theta, dist, ins, lW, lb, gW, gb, (float*)d_out, BN);
}